// XCA_80023830659441
// MI455X (gfx1250) — compile-verified
//
#include <hip/hip_runtime.h>
#include <hip/hip_bf16.h>
#include <math.h>

typedef __attribute__((ext_vector_type(2))) float v2f;
typedef __attribute__((ext_vector_type(8))) float v8f;

#define B_   16
#define C_   384
#define N_   1024
#define OC_  1152   // 3*C
#define NH_  8
#define HD_  48

__device__ __forceinline__ v8f wmma_f32(v2f a, v2f b, v8f c) {
  // V_WMMA_F32_16X16X4_F32 : D = A(16x4,f32) * B(4x16,f32) + C(16x16,f32)
  return __builtin_amdgcn_wmma_f32_16x16x4_f32(false, a, false, b, (short)0, c,
                                               false, false);
}

// ---------------------------------------------------------------------------
// K1: qkv = x_flat @ qkv_w^T + qkv_b
//   x is [B, C, N] in memory; logical row r = b*N + n, A[r][c] = x[b][c][n]
//   Output qkv: [B*N, 3C] row-major (workspace).
//   WG: 128 threads (4 waves) -> 16 rows x 128 cols; each wave owns 16x32
//   (two C tiles sharing each A fragment). K-step 32.
// ---------------------------------------------------------------------------
__global__ __launch_bounds__(128) void qkv_gemm(const float* __restrict__ x,
                                                const float* __restrict__ w,
                                                const float* __restrict__ bias,
                                                float* __restrict__ out) {
  __shared__ float As[32][16];       // As[k][m]
  __shared__ float Bs[32][129];      // Bs[k][j], odd pad
  const int tid  = threadIdx.x;
  const int lane = tid & 31;
  const int wave = tid >> 5;
  const int m0 = blockIdx.x * 16;         // global row (b*N + n)
  const int j0 = blockIdx.y * 128;        // output column base
  const int bat = m0 >> 10;               // / N_
  const int n0  = m0 & (N_ - 1);
  const float* xb = x + (size_t)bat * C_ * N_;

  const int m  = lane & 15;
  const int kb = (lane >> 4) << 1;
  const int mb = (lane >> 4) << 3;
  const int jw = wave * 32;               // this wave's 32-col slice in tile
  v8f acc0 = {}, acc1 = {};

  for (int k0 = 0; k0 < C_; k0 += 32) {
    {   // stage A: 16 x 32, coalesced over m
      int mm = tid & 15;
      int kk = tid >> 4;                  // 0..7
      #pragma unroll
      for (int p = 0; p < 4; ++p, kk += 8)
        As[kk][mm] = xb[(size_t)(k0 + kk) * N_ + n0 + mm];
    }
    {   // stage B: W^T[k][j] = w[j*C + k], coalesced over k
      int kk = tid & 31;
      int jj = tid >> 5;                  // 0..3
      #pragma unroll
      for (int p = 0; p < 32; ++p, jj += 4)
        Bs[kk][jj] = w[(size_t)(j0 + jj) * C_ + (k0 + kk)];
    }
    __syncthreads();
    #pragma unroll
    for (int ks = 0; ks < 32; ks += 4) {
      v2f a, b0, b1;
      a.x  = As[ks + kb][m];              a.y  = As[ks + kb + 1][m];
      b0.x = Bs[ks + kb][jw + m];         b0.y = Bs[ks + kb + 1][jw + m];
      b1.x = Bs[ks + kb][jw + 16 + m];    b1.y = Bs[ks + kb + 1][jw + 16 + m];
      acc0 = wmma_f32(a, b0, acc0);
      acc1 = wmma_f32(a, b1, acc1);
    }
    __syncthreads();
  }
  #pragma unroll
  for (int r = 0; r < 8; ++r) {
    int row  = m0 + mb + r;
    int col0 = j0 + jw + m;
    out[(size_t)row * OC_ + col0]      = acc0[r] + bias[col0];
    out[(size_t)row * OC_ + col0 + 16] = acc1[r] + bias[col0 + 16];
  }
}

// ---------------------------------------------------------------------------
// K2: L2-normalize q,k rows per head (in place); fold temperature into q.
// ---------------------------------------------------------------------------
__global__ void norm_qk(float* __restrict__ qkv, const float* __restrict__ temp) {
  int idx = blockIdx.x * blockDim.x + threadIdx.x;   // (row, head)
  if (idx >= B_ * N_ * NH_) return;
  int row = idx / NH_, h = idx % NH_;
  float* q = qkv + (size_t)row * OC_ + h * HD_;
  float* k = q + C_;
  float sq = 0.f, sk = 0.f;
  #pragma unroll 8
  for (int i = 0; i < HD_; ++i) { sq += q[i] * q[i]; sk += k[i] * k[i]; }
  float iq = temp[h] / fmaxf(sqrtf(sq), 1e-12f);
  float ik = 1.0f   / fmaxf(sqrtf(sk), 1e-12f);
  #pragma unroll 8
  for (int i = 0; i < HD_; ++i) { q[i] *= iq; k[i] *= ik; }
}

// ---------------------------------------------------------------------------
// K3: flash attention. One WG = 4 waves; each wave owns 32 query rows
// (two 16-row subtiles) -> WG covers 128 query rows of one (b,h).
// K/V tiles (16 x 48) staged in LDS, shared by all 4 waves; K fragments
// shared across both S accumulators; V fragments shared across both PV
// accumulator sets. Online softmax row stats match the C/D lane-half split.
// ---------------------------------------------------------------------------
__global__ __launch_bounds__(128) void flash_attn(const float* __restrict__ qkv,
                                                  float* __restrict__ ctx) {
  __shared__ float Ks[16][HD_ + 2];
  __shared__ float Vs[16][HD_ + 2];
  __shared__ float Ps[4][32][17];        // per-wave P scratch (D->A relayout)

  const int ntiles = N_ / 128;
  int t  = blockIdx.x % ntiles;
  int bh = blockIdx.x / ntiles;
  int h = bh % NH_, bat = bh / NH_;

  const int tid = threadIdx.x, lane = tid & 31, wave = tid >> 5;
  const int qrow0 = t * 128 + wave * 32;  // this wave's query row base
  const size_t rowbase = (size_t)bat * N_;
  const int qcol = h * HD_, kcol = C_ + h * HD_, vcol = 2 * C_ + h * HD_;

  const int m  = lane & 15;
  const int kb = (lane >> 4) << 1;
  const int mb = (lane >> 4) << 3;

  // Preload Q fragments for both 16-row subtiles (16x48 -> 12 k-steps each)
  v2f aq[2][12];
  #pragma unroll
  for (int u = 0; u < 2; ++u) {
    #pragma unroll
    for (int ks = 0; ks < 12; ++ks) {
      const float* qr = qkv + (rowbase + qrow0 + u * 16 + m) * (size_t)OC_
                            + qcol + ks * 4 + kb;
      aq[u][ks].x = qr[0]; aq[u][ks].y = qr[1];
    }
  }

  v8f o[2][3] = {{{}, {}, {}}, {{}, {}, {}}};
  float mst[2][8], lst[2][8];
  #pragma unroll
  for (int u = 0; u < 2; ++u)
    #pragma unroll
    for (int r = 0; r < 8; ++r) { mst[u][r] = -3.0e38f; lst[u][r] = 0.0f; }

  for (int j = 0; j < N_; j += 16) {
    // stage K,V tiles (coalesced, all 128 threads)
    for (int i = tid; i < 16 * HD_; i += 128) {
      int r = i / HD_, c = i % HD_;
      size_t base = (rowbase + j + r) * (size_t)OC_;
      Ks[r][c] = qkv[base + kcol + c];
      Vs[r][c] = qkv[base + vcol + c];
    }
    __syncthreads();

    // S = Q @ K^T : share each K fragment across both query subtiles
    v8f s[2] = {{}, {}};
    #pragma unroll
    for (int ks = 0; ks < 12; ++ks) {
      v2f bf;
      bf.x = Ks[m][ks * 4 + kb];
      bf.y = Ks[m][ks * 4 + kb + 1];
      s[0] = wmma_f32(aq[0][ks], bf, s[0]);
      s[1] = wmma_f32(aq[1][ks], bf, s[1]);
    }

    // online softmax per row (16-lane half butterflies), stash P in LDS
    #pragma unroll
    for (int u = 0; u < 2; ++u) {
      #pragma unroll
      for (int r = 0; r < 8; ++r) {
        float v = s[u][r];
        v = fmaxf(v, __shfl_xor(v, 1, 32));
        v = fmaxf(v, __shfl_xor(v, 2, 32));
        v = fmaxf(v, __shfl_xor(v, 4, 32));
        v = fmaxf(v, __shfl_xor(v, 8, 32));
        float mnew  = fmaxf(mst[u][r], v);
        float alpha = __expf(mst[u][r] - mnew);
        mst[u][r] = mnew;
        float p = __expf(s[u][r] - mnew);
        float rs = p;
        rs += __shfl_xor(rs, 1, 32);
        rs += __shfl_xor(rs, 2, 32);
        rs += __shfl_xor(rs, 4, 32);
        rs += __shfl_xor(rs, 8, 32);
        lst[u][r] = lst[u][r] * alpha + rs;
        o[u][0][r] *= alpha; o[u][1][r] *= alpha; o[u][2][r] *= alpha;
        Ps[wave][u * 16 + mb + r][m] = p;
      }
    }

    // O += P @ V : share each V fragment across both query subtiles
    #pragma unroll
    for (int ks = 0; ks < 4; ++ks) {
      v2f ap0, ap1, b0, b1, b2;
      ap0.x = Ps[wave][m][ks * 4 + kb];       ap0.y = Ps[wave][m][ks * 4 + kb + 1];
      ap1.x = Ps[wave][16 + m][ks * 4 + kb];  ap1.y = Ps[wave][16 + m][ks * 4 + kb + 1];
      b0.x = Vs[ks * 4 + kb][ 0 + m];  b0.y = Vs[ks * 4 + kb + 1][ 0 + m];
      b1.x = Vs[ks * 4 + kb][16 + m];  b1.y = Vs[ks * 4 + kb + 1][16 + m];
      b2.x = Vs[ks * 4 + kb][32 + m];  b2.y = Vs[ks * 4 + kb + 1][32 + m];
      o[0][0] = wmma_f32(ap0, b0, o[0][0]);
      o[0][1] = wmma_f32(ap0, b1, o[0][1]);
      o[0][2] = wmma_f32(ap0, b2, o[0][2]);
      o[1][0] = wmma_f32(ap1, b0, o[1][0]);
      o[1][1] = wmma_f32(ap1, b1, o[1][1]);
      o[1][2] = wmma_f32(ap1, b2, o[1][2]);
    }
    __syncthreads();
  }

  // normalize and store into ctx [B*N, C]
  #pragma unroll
  for (int u = 0; u < 2; ++u) {
    #pragma unroll
    for (int r = 0; r < 8; ++r) {
      float inv = 1.0f / lst[u][r];
      size_t base = (rowbase + qrow0 + u * 16 + mb + r) * (size_t)C_ + h * HD_;
      ctx[base +  0 + m] = o[u][0][r] * inv;
      ctx[base + 16 + m] = o[u][1][r] * inv;
      ctx[base + 32 + m] = o[u][2][r] * inv;
    }
  }
}

// ---------------------------------------------------------------------------
// K4: out^T = proj_w @ ctx^T + proj_b  (computes [B, C, N] directly ->
// final transpose to [B,C,H,W] is free, stores coalesced over n).
// WG: 16 channel rows x 128 bn cols; each wave owns 16x32 (two C tiles).
// ---------------------------------------------------------------------------
__global__ __launch_bounds__(128) void proj_gemm(const float* __restrict__ ctx,
                                                 const float* __restrict__ pw,
                                                 const float* __restrict__ pb,
                                                 float* __restrict__ out) {
  __shared__ float As[32][17];       // As[k][m] : proj_w rows
  __shared__ float Bs[32][129];      // Bs[k][j] : ctx^T cols
  const int tid = threadIdx.x, lane = tid & 31, wave = tid >> 5;
  const int m0 = blockIdx.x * 16;    // output channel rows
  const int c0 = blockIdx.y * 128;   // bn cols
  const int m  = lane & 15;
  const int kb = (lane >> 4) << 1;
  const int mb = (lane >> 4) << 3;
  const int jw = wave * 32;
  v8f acc0 = {}, acc1 = {};

  for (int k0 = 0; k0 < C_; k0 += 32) {
    {
      int kk = tid & 31, mm = tid >> 5;
      #pragma unroll
      for (int p = 0; p < 4; ++p, mm += 4)
        As[kk][mm] = pw[(size_t)(m0 + mm) * C_ + k0 + kk];
    }
    {
      int kk = tid & 31, jj = tid >> 5;
      #pragma unroll
      for (int p = 0; p < 32; ++p, jj += 4)
        Bs[kk][jj] = ctx[(size_t)(c0 + jj) * C_ + k0 + kk];
    }
    __syncthreads();
    #pragma unroll
    for (int ks = 0; ks < 32; ks += 4) {
      v2f a, b0, b1;
      a.x  = As[ks + kb][m];            a.y  = As[ks + kb + 1][m];
      b0.x = Bs[ks + kb][jw + m];       b0.y = Bs[ks + kb + 1][jw + m];
      b1.x = Bs[ks + kb][jw + 16 + m];  b1.y = Bs[ks + kb + 1][jw + 16 + m];
      acc0 = wmma_f32(a, b0, acc0);
      acc1 = wmma_f32(a, b1, acc1);
    }
    __syncthreads();
  }
  #pragma unroll
  for (int r = 0; r < 8; ++r) {
    int ch  = m0 + mb + r;
    float bsv = pb[ch];
    int colA = c0 + jw + m;
    int colB = colA + 16;
    out[(size_t)(colA >> 10) * C_ * N_ + (size_t)ch * N_ + (colA & (N_ - 1))] = acc0[r] + bsv;
    out[(size_t)(colB >> 10) * C_ * N_ + (size_t)ch * N_ + (colB & (N_ - 1))] = acc1[r] + bsv;
  }
}

// ---------------------------------------------------------------------------
extern "C" void kernel_launch(void* const* d_in, const int* in_sizes, int n_in,
                              void* d_out, int out_size, void* d_ws, size_t ws_size,
                              hipStream_t stream) {
  const float* x      = (const float*)d_in[0];
  const float* temp   = (const float*)d_in[1];
  const float* qkv_w  = (const float*)d_in[2];
  const float* qkv_b  = (const float*)d_in[3];
  const float* proj_w = (const float*)d_in[4];
  const float* proj_b = (const float*)d_in[5];
  float* out = (float*)d_out;

  float* qkv = (float*)d_ws;                        // [B*N, 3C]  75.5 MB
  float* ctx = qkv + (size_t)B_ * N_ * OC_;         // [B*N, C]   25.2 MB

  dim3 g1(B_ * N_ / 16, OC_ / 128);                 // (1024, 9)
  qkv_gemm<<<g1, 128, 0, stream>>>(x, qkv_w, qkv_b, qkv);

  int pairs = B_ * N_ * NH_;                        // 131072
  norm_qk<<<(pairs + 255) / 256, 256, 0, stream>>>(qkv, temp);

  flash_attn<<<B_ * NH_ * (N_ / 128), 128, 0, stream>>>(qkv, ctx);  // 1024 WGs

  dim3 g4(C_ / 16, (B_ * N_) / 128);                // (24, 128)
  proj_gemm<<<g4, 128, 0, stream>>>(ctx, proj_w, proj_b, out);
}